// LightGCN_17265768530180
// MI455X (gfx1250) — compile-verified
//
#include <hip/hip_runtime.h>
#include <hip/hip_bf16.h>

typedef __attribute__((ext_vector_type(2))) float v2f;
typedef __attribute__((ext_vector_type(8))) float v8f;

#define N_USERS   100000
#define N_ITEMS   100000
#define N_NODES   (N_USERS + N_ITEMS)
#define D_MODEL   64
#define BATCH     2048
#define HIST      50
#define N_EDGES   6400000
#define N_TILES_N (N_ITEMS / 16)          // 6250 exact
#define GEMM_GRID_X 32
#define WAVES_PER_BLOCK 8
#define N_WAVECOLS (GEMM_GRID_X * WAVES_PER_BLOCK)  // 256

// ---------------------------------------------------------------------------
// 1) x0 = concat(user_emb, item_emb[1:]);  acc = x0
// ---------------------------------------------------------------------------
__global__ void lgcn_init_x(const float* __restrict__ user_emb,
                            const float* __restrict__ item_emb,
                            float* __restrict__ X, float* __restrict__ ACC) {
    long long i = (long long)blockIdx.x * blockDim.x + threadIdx.x;
    if (i >= (long long)N_NODES * D_MODEL) return;
    int n = (int)(i >> 6);
    int d = (int)(i & 63);
    float v = (n < N_USERS) ? user_emb[i]
                            : item_emb[(long long)(n - N_USERS + 1) * D_MODEL + d];
    X[i]   = v;
    ACC[i] = v;
}

// ---------------------------------------------------------------------------
// 2) SpMM scatter: Y[row] += val * X[col]   (Y pre-zeroed)
//    16 threads per edge, float4 per thread -> coalesced 256B gathers
// ---------------------------------------------------------------------------
__global__ void lgcn_spmm(const int* __restrict__ arow, const int* __restrict__ acol,
                          const float* __restrict__ aval,
                          const float* __restrict__ X, float* __restrict__ Y) {
    long long tid = (long long)blockIdx.x * blockDim.x + threadIdx.x;
    if (tid >= (long long)N_EDGES * 16) return;
    int e = (int)(tid >> 4);
    int q = (int)(tid & 15) * 4;
    float v = aval[e];
    const float4 x = *(const float4*)(X + (long long)acol[e] * D_MODEL + q);
    float* yd = Y + (long long)arow[e] * D_MODEL + q;
    atomicAdd(yd + 0, v * x.x);
    atomicAdd(yd + 1, v * x.y);
    atomicAdd(yd + 2, v * x.z);
    atomicAdd(yd + 3, v * x.w);
}

// ---------------------------------------------------------------------------
// 3) ACC += Y  /  ACC *= 0.25
// ---------------------------------------------------------------------------
__global__ void lgcn_acc_add(float* __restrict__ ACC, const float* __restrict__ Y) {
    long long i = (long long)blockIdx.x * blockDim.x + threadIdx.x;
    if (i < (long long)N_NODES * D_MODEL) ACC[i] += Y[i];
}
__global__ void lgcn_scale(float* __restrict__ ACC) {
    long long i = (long long)blockIdx.x * blockDim.x + threadIdx.x;
    if (i < (long long)N_NODES * D_MODEL) ACC[i] *= 0.25f;
}

// ---------------------------------------------------------------------------
// 4) pred[b] = user_latent[user_seqs[b]] + mean_nonpad(item_emb[his[b]])
//    (item_emb row 0 is all-zero, so summing pads is harmless)
// ---------------------------------------------------------------------------
__global__ void lgcn_pred(const float* __restrict__ ACC, const float* __restrict__ item_emb,
                          const int* __restrict__ user_seqs, const int* __restrict__ his,
                          float* __restrict__ pred) {
    int b = blockIdx.x;
    int d = threadIdx.x;                       // 64 threads
    float s = 0.f;
    int cnt = 0;
    #pragma unroll 1
    for (int t = 0; t < HIST; ++t) {
        int it = his[b * HIST + t];
        cnt += (it != 0);
        s += item_emb[(long long)it * D_MODEL + d];
    }
    pred[b * D_MODEL + d] =
        ACC[(long long)user_seqs[b] * D_MODEL + d] + s / (float)cnt;
}

// ---------------------------------------------------------------------------
// 4b) labsc[b] = dot(pred[b], item_latent[next_items[b]-1])
//     Tiny side computation -> keeps all label logic out of the GEMM hot loop.
// ---------------------------------------------------------------------------
__global__ void lgcn_label_score(const float* __restrict__ pred,
                                 const float* __restrict__ item,
                                 const int* __restrict__ next_items,
                                 float* __restrict__ labsc) {
    __shared__ float sh[64];
    int b = blockIdx.x;
    int d = threadIdx.x;                       // 64 threads
    int lab = next_items[b] - 1;
    sh[d] = pred[b * D_MODEL + d] * item[(long long)lab * D_MODEL + d];
    __syncthreads();
    for (int s = 32; s > 0; s >>= 1) {
        if (d < s) sh[d] += sh[d + s];
        __syncthreads();
    }
    if (d == 0) labsc[b] = sh[0];
}

// ---------------------------------------------------------------------------
// 5) scores = pred @ item_latent^T fused with lane-local online log-sum-exp.
//    V_WMMA_F32_16X16X4_F32, K=64 as 16 back-to-back WMMAs per tile; B tiles
//    double-buffered in registers with a manual 2x ping-pong (no reg copies).
//    __launch_bounds__(256,1) buys the VGPR budget for both B buffers + A.
//    Layouts (ISA 7.12.2):
//      A 16x4 f32: VGPR0 = {K=0 | lanes0-15, K=2 | lanes16-31}, VGPR1 = {K=1|K=3}
//      C/D 16x16 f32: VGPR r = {row r | lanes0-15, row r+8 | lanes16-31}
//    Each lane owns one column-slot of 8 rows -> running (max,sum) kept in
//    registers; cross-lane merge happens once at the end (width-16 butterfly).
// ---------------------------------------------------------------------------
__device__ __forceinline__ void lgcn_load_btile(v2f* __restrict__ b,
                                                const float* __restrict__ item,
                                                int t, int l16, int koff) {
    const float* brow = item + (long long)(t * 16 + l16) * D_MODEL + koff;
    #pragma unroll
    for (int ks = 0; ks < 16; ++ks) {
        b[ks][0] = brow[ks * 4 + 0];
        b[ks][1] = brow[ks * 4 + 1];
    }
}

__device__ __forceinline__ void lgcn_tile_mma_lse(const v2f* __restrict__ a,
                                                  const v2f* __restrict__ b,
                                                  float* __restrict__ run_mx,
                                                  float* __restrict__ run_sm) {
    v8f c = {};
    #pragma unroll
    for (int ks = 0; ks < 16; ++ks)
        c = __builtin_amdgcn_wmma_f32_16x16x4_f32(
                false, a[ks], false, b[ks], (short)0, c, false, false);
    #pragma unroll
    for (int i = 0; i < 8; ++i) {
        float s  = c[i];
        float nm = fmaxf(run_mx[i], s);
        run_sm[i] = run_sm[i] * __expf(run_mx[i] - nm) + __expf(s - nm);
        run_mx[i] = nm;
    }
}

__global__ void __launch_bounds__(256, 1)
lgcn_gemm_lse(const float* __restrict__ pred,   // [BATCH][64]
              const float* __restrict__ item,   // [N_ITEMS][64]
              float* __restrict__ part_mx,      // [BATCH][N_WAVECOLS]
              float* __restrict__ part_sm) {    // [BATCH][N_WAVECOLS]
    const int m_base = blockIdx.y * 16;
    const int wave   = threadIdx.x >> 5;
    const int lane   = threadIdx.x & 31;
    const int half   = lane >> 4;            // 0 or 1
    const int l16    = lane & 15;
    const int koff   = half * 2;
    const int wcid   = blockIdx.x * WAVES_PER_BLOCK + wave;

    // A tile resident in registers: 16 k-steps x v2f
    v2f a_reg[16];
    {
        const float* ap = pred + (m_base + l16) * D_MODEL + koff;
        #pragma unroll
        for (int ks = 0; ks < 16; ++ks) {
            a_reg[ks][0] = ap[ks * 4 + 0];
            a_reg[ks][1] = ap[ks * 4 + 1];
        }
    }

    float run_mx[8], run_sm[8];
    #pragma unroll
    for (int i = 0; i < 8; ++i) { run_mx[i] = -__builtin_inff(); run_sm[i] = 0.f; }

    // manual ping-pong double buffer: no register copies between stages
    v2f b0[16], b1[16];
    int t = wcid;
    if (t < N_TILES_N) lgcn_load_btile(b0, item, t, l16, koff);
    while (t < N_TILES_N) {
        int tn = t + N_WAVECOLS;
        if (tn < N_TILES_N) lgcn_load_btile(b1, item, tn, l16, koff);
        lgcn_tile_mma_lse(a_reg, b0, run_mx, run_sm);
        t = tn;
        if (t >= N_TILES_N) break;
        tn = t + N_WAVECOLS;
        if (tn < N_TILES_N) lgcn_load_btile(b0, item, tn, l16, koff);
        lgcn_tile_mma_lse(a_reg, b1, run_mx, run_sm);
        t = tn;
    }

    // one-time width-16 merge butterfly across the half-wave
    #pragma unroll
    for (int i = 0; i < 8; ++i) {
        #pragma unroll
        for (int d = 8; d >= 1; d >>= 1) {
            float omx = __shfl_xor(run_mx[i], d, 16);
            float osm = __shfl_xor(run_sm[i], d, 16);
            float nm  = fmaxf(run_mx[i], omx);
            run_sm[i] = run_sm[i] * __expf(run_mx[i] - nm) + osm * __expf(omx - nm);
            run_mx[i] = nm;
        }
    }
    if (l16 == 0) {
        #pragma unroll
        for (int i = 0; i < 8; ++i) {
            int row = m_base + i + 8 * half;
            part_mx[(long long)row * N_WAVECOLS + wcid] = run_mx[i];
            part_sm[(long long)row * N_WAVECOLS + wcid] = run_sm[i];
        }
    }
}

// ---------------------------------------------------------------------------
// 6) loss = mean(lse_row - label_score_row); single block, deterministic
// ---------------------------------------------------------------------------
__global__ void lgcn_loss(const float* __restrict__ part_mx,
                          const float* __restrict__ part_sm,
                          const float* __restrict__ labsc,
                          float* __restrict__ out) {
    __shared__ float red[1024];
    float acc = 0.f;
    for (int row = threadIdx.x; row < BATCH; row += 1024) {
        const float* pm = part_mx + (long long)row * N_WAVECOLS;
        const float* ps = part_sm + (long long)row * N_WAVECOLS;
        float M = -__builtin_inff();
        for (int j = 0; j < N_WAVECOLS; ++j) M = fmaxf(M, pm[j]);
        float S = 0.f;
        for (int j = 0; j < N_WAVECOLS; ++j) S += ps[j] * __expf(pm[j] - M);
        acc += (M + __logf(S)) - labsc[row];
    }
    red[threadIdx.x] = acc;
    __syncthreads();
    for (int s = 512; s > 0; s >>= 1) {
        if (threadIdx.x < (unsigned)s) red[threadIdx.x] += red[threadIdx.x + s];
        __syncthreads();
    }
    if (threadIdx.x == 0) out[0] = red[0] / (float)BATCH;
}

// ---------------------------------------------------------------------------
extern "C" void kernel_launch(void* const* d_in, const int* in_sizes, int n_in,
                              void* d_out, int out_size, void* d_ws, size_t ws_size,
                              hipStream_t stream) {
    const int*   user_seqs  = (const int*)  d_in[0];
    const int*   his_seqs   = (const int*)  d_in[1];
    const int*   next_items = (const int*)  d_in[2];
    const int*   adj_row    = (const int*)  d_in[3];
    const int*   adj_col    = (const int*)  d_in[4];
    const float* adj_val    = (const float*)d_in[5];
    const float* user_emb   = (const float*)d_in[6];
    const float* item_emb   = (const float*)d_in[7];

    const size_t NODE_FLOATS = (size_t)N_NODES * D_MODEL;        // 12.8M
    char* ws = (char*)d_ws;
    float* X0      = (float*)(ws);                                // 51.2 MB
    float* X1      = (float*)(ws + NODE_FLOATS * 4);              // 51.2 MB
    float* ACC     = (float*)(ws + NODE_FLOATS * 8);              // 51.2 MB
    float* PRED    = (float*)(ws + NODE_FLOATS * 12);             // 0.5 MB
    float* PART_MX = PRED + (size_t)BATCH * D_MODEL;
    float* PART_SM = PART_MX + (size_t)BATCH * N_WAVECOLS;
    float* LABSC   = PART_SM + (size_t)BATCH * N_WAVECOLS;

    // init x/acc
    {
        long long n = (long long)N_NODES * D_MODEL;
        lgcn_init_x<<<(unsigned)((n + 255) / 256), 256, 0, stream>>>(user_emb, item_emb, X0, ACC);
    }
    // 3 propagation layers, ping-pong X0<->X1
    {
        long long spmm_threads = (long long)N_EDGES * 16;
        unsigned  spmm_blocks  = (unsigned)((spmm_threads + 255) / 256);
        long long n = (long long)N_NODES * D_MODEL;
        unsigned  nb = (unsigned)((n + 255) / 256);
        float* src = X0;
        float* dst = X1;
        for (int layer = 0; layer < 3; ++layer) {
            hipMemsetAsync(dst, 0, NODE_FLOATS * 4, stream);
            lgcn_spmm<<<spmm_blocks, 256, 0, stream>>>(adj_row, adj_col, adj_val, src, dst);
            lgcn_acc_add<<<nb, 256, 0, stream>>>(ACC, dst);
            float* t = src; src = dst; dst = t;
        }
        lgcn_scale<<<nb, 256, 0, stream>>>(ACC);
    }
    const float* ITEM_LATENT = ACC + (size_t)N_USERS * D_MODEL;
    // pred rows
    lgcn_pred<<<BATCH, D_MODEL, 0, stream>>>(ACC, item_emb, user_seqs, his_seqs, PRED);
    // label scores (side channel, out of the GEMM hot loop)
    lgcn_label_score<<<BATCH, D_MODEL, 0, stream>>>(PRED, ITEM_LATENT, next_items, LABSC);
    // fused WMMA GEMM + online logsumexp over 100k items
    {
        dim3 grid(GEMM_GRID_X, BATCH / 16);
        lgcn_gemm_lse<<<grid, 32 * WAVES_PER_BLOCK, 0, stream>>>(
            PRED, ITEM_LATENT, PART_MX, PART_SM);
    }
    // final scalar loss
    lgcn_loss<<<1, 1024, 0, stream>>>(PART_MX, PART_SM, LABSC, (float*)d_out);
    (void)in_sizes; (void)n_in; (void)out_size; (void)ws_size;
}